// SimpleRNN_8589934592513
// MI455X (gfx1250) — compile-verified
//
#include <hip/hip_runtime.h>
#include <math.h>

// SimpleRNN on MI455X (gfx1250):
//   Xe = emb[X]; Xproj = Xe @ W_xh^T + b_xh
//   h_{t+1} = tanh(h_t @ W_hh^T + b_hh + Xproj[:,t,:])
// All GEMMs run on the wave32 WMMA pipe with exact-f32 V_WMMA_F32_16X16X4_F32.
// Weights are pre-transposed AND K-pair-interleaved so each WMMA B operand is
// one contiguous global_load_b64 per lane.

#define VOCAB  32000
#define EMB    256
#define HID    512
#define NB     256
#define SEQ    512

typedef float v2f __attribute__((ext_vector_type(2)));
typedef float v8f __attribute__((ext_vector_type(8)));

#define LH_STRIDE (HID + 4)   // pad so 16x4 f32 A-tile fetch is LDS-bank-conflict-free
#define LE_STRIDE (EMB + 4)
#define PKROW     (2 * HID)   // floats per packed K-pair row

// ---------------------------------------------------------------------------
// One-time prep: build K-pair-interleaved transposed weights
//   Wpk[(k>>1)*PKROW + j*2 + (k&1)] = W[j][k]
// so the step kernel's B operand (b.x = W^T[k][j], b.y = W^T[k+1][j]) is a
// single float2 load. Also zero the initial hidden state (lives in d_out).
// ---------------------------------------------------------------------------
__global__ __launch_bounds__(256) void rnn_init_kernel(
    const float* __restrict__ Whh, const float* __restrict__ Wxh,
    float* __restrict__ WhhP, float* __restrict__ WxhP,
    float* __restrict__ h0)
{
    int i = blockIdx.x * 256 + threadIdx.x;
    if (i < HID * HID) {
        int j = i / HID, k = i % HID;          // Whh is (HID out, HID in) row-major
        WhhP[(k >> 1) * PKROW + j * 2 + (k & 1)] = Whh[i];
    } else if (i < HID * HID + HID * EMB) {
        int r = i - HID * HID;
        int j = r / EMB, e = r % EMB;          // Wxh is (HID, EMB) row-major
        WxhP[(e >> 1) * PKROW + j * 2 + (e & 1)] = Wxh[r];
    } else if (i < HID * HID + HID * EMB + NB * HID) {
        h0[i - HID * HID - HID * EMB] = 0.0f;  // h_0 = 0
    }
}

// ---------------------------------------------------------------------------
// One recurrence step, fused with the embedding gather + input projection:
//   h_out[n][j] = tanh( sum_e emb[X[n][t]][e]*Wxh[j][e]
//                     + sum_k h_in[n][k] *Whh[j][k] + b_hh[j] + b_xh[j] )
// Grid: 64 blocks x 256 threads. Block = (ntile, jgroup); the 8 waves of a
// block share one 16-row A-panel (staged in LDS) and cover 8 j-tiles.
// Each wave: one 16x16 f32 tile via v_wmma_f32_16x16x4_f32 over K=256+512.
// ---------------------------------------------------------------------------
__global__ __launch_bounds__(256) void rnn_step_kernel(
    const int*   __restrict__ X,
    const float* __restrict__ emb,
    const float* __restrict__ WhhP,
    const float* __restrict__ WxhP,
    const float* __restrict__ b_hh,
    const float* __restrict__ b_xh,
    const float* __restrict__ h_in,
    float*       __restrict__ h_out,
    int t)
{
    __shared__ float lhs_h[16 * LH_STRIDE];  // h rows n0..n0+15    (33 KB)
    __shared__ float lhs_e[16 * LE_STRIDE];  // emb rows (gathered) (17 KB)
    __shared__ int   toks[16];

    const int tid   = threadIdx.x;
    const int ntile = blockIdx.x >> 2;   // 0..15  -> rows n0..n0+15
    const int jg    = blockIdx.x & 3;    // 0..3   -> j-tile group
    const int n0    = ntile * 16;

    // tokens for this row panel
    if (tid < 16) toks[tid] = X[(n0 + tid) * SEQ + t];
    __syncthreads();

    // Stage A-panels into LDS (coalesced float4 copies, padded rows).
    {
        const float* src = h_in + (size_t)n0 * HID;
        for (int i = tid; i < 16 * (HID / 4); i += 256) {
            int r = i >> 7, c = i & 127;                     // 128 float4 per row
            float4 v = ((const float4*)(src + r * HID))[c];
            float* dst = &lhs_h[r * LH_STRIDE + c * 4];
            dst[0] = v.x; dst[1] = v.y; dst[2] = v.z; dst[3] = v.w;
        }
        for (int i = tid; i < 16 * (EMB / 4); i += 256) {
            int r = i >> 6, c = i & 63;                      // 64 float4 per row
            const float* es = emb + (size_t)toks[r] * EMB;
            float4 v = ((const float4*)es)[c];
            float* dst = &lhs_e[r * LE_STRIDE + c * 4];
            dst[0] = v.x; dst[1] = v.y; dst[2] = v.z; dst[3] = v.w;
        }
    }
    __syncthreads();

    const int wave = tid >> 5;
    const int lane = tid & 31;
    const int lo   = lane & 15;          // A row / B column within tile
    const int hi   = lane >> 4;          // K sub-offset select (0 -> K0..1, 1 -> K2..3)
    const int j0   = (jg * 8 + wave) * 16;

    v8f acc0 = {}; v8f acc1 = {};        // two accumulators: break WMMA RAW chain

    // --- input-projection contribution: K = EMB ---
    {
        const float* abase = &lhs_e[lo * LE_STRIDE + hi * 2];
        // packed B base for this lane: row pair (k/2 + hi), column j0+lo
        const float* bbase = WxhP + (size_t)hi * PKROW + (j0 + lo) * 2;
        #pragma unroll 4
        for (int k0 = 0; k0 < EMB; k0 += 8) {
            v2f a0 = *(const v2f*)(abase + k0);
            v2f b0 = *(const v2f*)(bbase + (size_t)(k0 >> 1) * PKROW);
            acc0 = __builtin_amdgcn_wmma_f32_16x16x4_f32(
                       false, a0, false, b0, (short)0, acc0, false, false);

            v2f a1 = *(const v2f*)(abase + k0 + 4);
            v2f b1 = *(const v2f*)(bbase + (size_t)((k0 >> 1) + 2) * PKROW);
            acc1 = __builtin_amdgcn_wmma_f32_16x16x4_f32(
                       false, a1, false, b1, (short)0, acc1, false, false);
        }
    }

    // --- recurrent contribution: K = HID ---
    {
        const float* abase = &lhs_h[lo * LH_STRIDE + hi * 2];
        const float* bbase = WhhP + (size_t)hi * PKROW + (j0 + lo) * 2;
        #pragma unroll 4
        for (int k0 = 0; k0 < HID; k0 += 8) {
            v2f a0 = *(const v2f*)(abase + k0);
            v2f b0 = *(const v2f*)(bbase + (size_t)(k0 >> 1) * PKROW);
            acc0 = __builtin_amdgcn_wmma_f32_16x16x4_f32(
                       false, a0, false, b0, (short)0, acc0, false, false);

            v2f a1 = *(const v2f*)(abase + k0 + 4);
            v2f b1 = *(const v2f*)(bbase + (size_t)((k0 >> 1) + 2) * PKROW);
            acc1 = __builtin_amdgcn_wmma_f32_16x16x4_f32(
                       false, a1, false, b1, (short)0, acc1, false, false);
        }
    }

    v8f acc = acc0 + acc1;

    // --- bias + tanh + store ---
    const int   col  = j0 + lo;
    const float bias = b_hh[col] + b_xh[col];
    #pragma unroll
    for (int v = 0; v < 8; v++) {
        // C/D layout: VGPR v, lanes 0-15 -> M=v, lanes 16-31 -> M=v+8, N=lane&15
        int row = n0 + v + hi * 8;
        h_out[(size_t)row * HID + col] = tanhf(acc[v] + bias);
    }
}

// ---------------------------------------------------------------------------
extern "C" void kernel_launch(void* const* d_in, const int* in_sizes, int n_in,
                              void* d_out, int out_size, void* d_ws, size_t ws_size,
                              hipStream_t stream) {
    const int*   X    = (const int*)  d_in[0];
    const float* emb  = (const float*)d_in[1];
    const float* Whh  = (const float*)d_in[2];
    const float* bhh  = (const float*)d_in[3];
    const float* Wxh  = (const float*)d_in[4];
    const float* bxh  = (const float*)d_in[5];
    float* out = (float*)d_out;
    float* ws  = (float*)d_ws;

    float* WhhP = ws;                                  // 512*512 floats (1 MB)
    float* WxhP = ws + HID * HID;                      // 256*512 floats (0.5 MB)
    float* hbuf = ws + HID * HID + EMB * HID;          // 256*512 floats (0.5 MB)

    // pack weights + zero h0 (h0 lives in d_out so step 0 reads zeros)
    {
        int total = HID * HID + HID * EMB + NB * HID;  // 524288
        rnn_init_kernel<<<(total + 255) / 256, 256, 0, stream>>>(
            Whh, Wxh, WhhP, WxhP, out);
    }

    // 512 sequential steps; ping-pong between d_out and ws so that the final
    // step (t=511, odd) writes the last hidden state directly into d_out.
    for (int t = 0; t < SEQ; t++) {
        const float* hin  = (t & 1) ? hbuf : out;
        float*       hout = (t & 1) ? out  : hbuf;
        rnn_step_kernel<<<64, 256, 0, stream>>>(
            X, emb, WhhP, WxhP, bhh, bxh, hin, hout, t);
    }
}